// NTMMemory_6416681140878
// MI455X (gfx1250) — compile-verified
//
#include <hip/hip_runtime.h>
#include <math.h>

// NTM memory update for MI455X (gfx1250, wave32).
// One workgroup (256 thr = 8 waves) per batch. The whole 1024x64 f32 memory
// slab (256KB) is DMA'd into the 320KB WGP LDS by the Tensor Data Mover with
// hardware row padding (pitch 68 floats -> bank-conflict-free WMMA A loads).
// All phases run from LDS; HBM is touched exactly once per element.
// Dot products + row norms use V_WMMA_F32_16X16X4_F32 (exact f32 math).

typedef float v2f __attribute__((ext_vector_type(2)));
typedef float v8f __attribute__((ext_vector_type(8)));
typedef unsigned int u32x4 __attribute__((ext_vector_type(4)));
typedef int i32x4 __attribute__((ext_vector_type(4)));
typedef int i32x8 __attribute__((ext_vector_type(8)));

#define NROWS 1024
#define MCOLS 64
#define PITCH 68          // floats; 272B rows (TDM pad: +4 DWORDs every 64 DWORDs)
#define NBATCH 1024
#define EPSF 1e-16f
#define COS_EPSF 1e-8f

__device__ __forceinline__ float block_reduce_sum(float v, float* wscr, int lane, int wid) {
  #pragma unroll
  for (int off = 16; off; off >>= 1) v += __shfl_xor(v, off, 32);
  __syncthreads();                 // protect wscr reuse from previous reduction
  if (lane == 0) wscr[wid] = v;
  __syncthreads();
  float s = 0.f;
  #pragma unroll
  for (int i = 0; i < 8; ++i) s += wscr[i];
  return s;
}

__device__ __forceinline__ float block_reduce_max(float v, float* wscr, int lane, int wid) {
  #pragma unroll
  for (int off = 16; off; off >>= 1) v = fmaxf(v, __shfl_xor(v, off, 32));
  __syncthreads();
  if (lane == 0) wscr[wid] = v;
  __syncthreads();
  float m = wscr[0];
  #pragma unroll
  for (int i = 1; i < 8; ++i) m = fmaxf(m, wscr[i]);
  return m;
}

__global__ __launch_bounds__(256) void ntm_memory_kernel(
    const float* __restrict__ memory, const float* __restrict__ k,
    const float* __restrict__ beta_p, const float* __restrict__ g_p,
    const float* __restrict__ s_p,    const float* __restrict__ gamma_p,
    const float* __restrict__ w_prev, const float* __restrict__ e_p,
    const float* __restrict__ a_p,    float* __restrict__ out)
{
  extern __shared__ float smem[];
  float* tile = smem;                  // [1024 * 68] memory slab (padded pitch)
  float* dotA = tile + NROWS * PITCH;  // [1024] dot -> logit -> exp -> w_hat^g -> w
  float* auxA = dotA + NROWS;          // [1024] norm2 -> wg
  float* kb   = auxA + NROWS;          // [64] k + EPS
  float* eb   = kb + 64;               // [64]
  float* ab   = eb + 64;               // [64]
  float* racc = ab + 64;               // [64] read accumulator
  float* wscr = racc + 64;             // [32] wave reduction scratch

  const int b    = blockIdx.x;
  const int tid  = threadIdx.x;
  const int lane = tid & 31;
  const int wid  = tid >> 5;
  const int half = lane >> 4;          // 0: lanes 0-15, 1: lanes 16-31
  const int l    = lane & 15;

  const float beta  = beta_p[b];
  const float gg    = g_p[b];
  const float s0    = s_p[3*b + 0];
  const float s1    = s_p[3*b + 1];
  const float s2    = s_p[3*b + 2];
  const float gamma = gamma_p[b];

  // ---- stage 0: small vectors + zero accumulators ----
  if (tid < 64)        kb[tid]         = k[b*64 + tid] + EPSF;
  else if (tid < 128)  eb[tid - 64]    = e_p[b*64 + tid - 64];
  else if (tid < 192)  ab[tid - 128]   = a_p[b*64 + tid - 128];
  else                 racc[tid - 192] = 0.f;

  // ---- stage 1: DMA memory[b] (1024x64 f32) into LDS via the TDM ----
#if __has_builtin(__builtin_amdgcn_tensor_load_to_lds)
  if (wid == 0) {
    const unsigned int lds_base = (unsigned int)(uintptr_t)tile;  // addr[31:0] = LDS offset
    const unsigned long long ga =
        (unsigned long long)(uintptr_t)(memory + (size_t)b * NROWS * MCOLS);
    // D# group 0: count=1, lds_addr, 57-bit global addr, type=2 ("image")
    u32x4 g0;
    g0[0] = 1u;
    g0[1] = lds_base;
    g0[2] = (unsigned int)(ga & 0xffffffffull);
    g0[3] = (unsigned int)((ga >> 32) & 0x01ffffffull) | (2u << 30);
    // D# group 1: data_size=4B(code2), pad_enable, pad_interval=64dw(code5),
    // pad_amount=4dw(code3) -> LDS pitch 68 floats. 2D tensor 64x1024, tile=whole.
    i32x8 g1;
    g1[0] = (int)((2u << 16) | (1u << 20) | (5u << 22) | (3u << 25));
    g1[1] = (int)((unsigned)(MCOLS & 0xffff) << 16);                 // tensor_dim0 lo
    g1[2] = (int)(((unsigned)MCOLS >> 16) |
                  ((unsigned)(NROWS & 0xffff) << 16));               // dim0 hi, dim1 lo
    g1[3] = (int)(((unsigned)NROWS >> 16) |
                  ((unsigned)MCOLS << 16));                          // dim1 hi, tile_dim0
    g1[4] = (int)((unsigned)NROWS & 0xffff);                         // tile_dim1; tile_dim2=0
    g1[5] = (int)MCOLS;                                              // tensor_dim0_stride lo
    g1[6] = 0;                                                       // stride0 hi, stride1 lo
    g1[7] = 0;
    i32x4 gz = {0, 0, 0, 0};
#if defined(__clang_major__) && (__clang_major__ >= 23)
    i32x8 gz8 = {0, 0, 0, 0, 0, 0, 0, 0};
    __builtin_amdgcn_tensor_load_to_lds(g0, g1, gz, gz, gz8, 0);
#else
    __builtin_amdgcn_tensor_load_to_lds(g0, g1, gz, gz, 0);
#endif
    __builtin_amdgcn_s_wait_tensorcnt(0);   // TENSORcnt==0: tile resident in LDS
  }
#else
  // Fallback: manual staging through VGPRs
  {
    const float4* gmem = (const float4*)(memory + (size_t)b * NROWS * MCOLS);
    for (int idx = tid; idx < NROWS * MCOLS / 4; idx += 256) {
      int row = idx >> 4;
      int c4  = (idx & 15) * 4;
      float4 v = gmem[idx];
      *(float4*)(tile + row * PITCH + c4) = v;
    }
  }
#endif
  __syncthreads();

  // ---- k norm (uniform broadcast reads) ----
  float kn2 = 0.f;
  #pragma unroll
  for (int m = 0; m < 64; ++m) kn2 += kb[m] * kb[m];
  const float kden = fmaxf(sqrtf(kn2), COS_EPSF);

  // ---- stage 2: dot[n] and norm2[n] via V_WMMA_F32_16X16X4_F32 ----
  // A layout (ISA 7.12.2): lanes 0-15 row M=l K0/K1, lanes 16-31 row M=l K2/K3.
  // B mirrors: VGPR0 = K0 (lanes 0-15) / K2 (lanes 16-31), VGPR1 = K1/K3.
  // B_dot: every column = k chunk  -> D[i][*] = dot(row i, k) partial.
  // B_gram = A registers exactly   -> D = A*A^T, diagonal = row norms^2.
  v2f karr[16];
  #pragma unroll
  for (int ch = 0; ch < 16; ++ch) {
    karr[ch].x = kb[4*ch + 2*half + 0];
    karr[ch].y = kb[4*ch + 2*half + 1];
  }
  for (int gi = 0; gi < 8; ++gi) {
    const int rowBase = wid * 128 + gi * 16;
    const float* rp = tile + (rowBase + l) * PITCH + 2 * half;
    v8f cd = {};
    v8f cg = {};
    #pragma unroll
    for (int ch = 0; ch < 16; ++ch) {
      v2f A;
      A.x = rp[4*ch + 0] + EPSF;
      A.y = rp[4*ch + 1] + EPSF;
      cd = __builtin_amdgcn_wmma_f32_16x16x4_f32(false, A, false, karr[ch],
                                                 (short)0, cd, false, false);
      cg = __builtin_amdgcn_wmma_f32_16x16x4_f32(false, A, false, A,
                                                 (short)0, cg, false, false);
    }
    // D layout: VGPR r = row M=r (lanes 0-15, N=lane) / M=r+8 (lanes 16-31, N=lane-16)
    if (l == 0) {            // column 0 carries the dot value
      #pragma unroll
      for (int r = 0; r < 8; ++r) dotA[rowBase + half*8 + r] = cd[r];
    }
    #pragma unroll
    for (int r = 0; r < 8; ++r) {    // Gram diagonal D[i][i]
      if (half == 0 && l == r)     auxA[rowBase + r]     = cg[r];
      if (half == 1 && l == r + 8) auxA[rowBase + 8 + r] = cg[r];
    }
  }
  __syncthreads();

  // ---- stage 3: logits -> softmax -> interpolate -> shift -> sharpen ----
  float lg[4];
  #pragma unroll
  for (int j = 0; j < 4; ++j) {
    const int n = tid + 256 * j;
    const float nd = fmaxf(sqrtf(auxA[n]), COS_EPSF) * kden;
    lg[j] = beta * (dotA[n] / nd);
  }
  float lmax = fmaxf(fmaxf(lg[0], lg[1]), fmaxf(lg[2], lg[3]));
  const float bmax = block_reduce_max(lmax, wscr, lane, wid);

  float ex[4]; float lsum = 0.f;
  #pragma unroll
  for (int j = 0; j < 4; ++j) { ex[j] = __expf(lg[j] - bmax); lsum += ex[j]; }
  const float ssum = block_reduce_sum(lsum, wscr, lane, wid);
  const float sinv = 1.f / ssum;

  #pragma unroll
  for (int j = 0; j < 4; ++j) {
    const int n = tid + 256 * j;
    auxA[n] = gg * (ex[j] * sinv) + (1.f - gg) * w_prev[(size_t)b * NROWS + n];
  }
  __syncthreads();   // wg fully written before circular shift

  float wv[4]; float psum = 0.f;
  #pragma unroll
  for (int j = 0; j < 4; ++j) {
    const int n = tid + 256 * j;
    const float wh = s0 * auxA[(n + NROWS - 1) & (NROWS - 1)]
                   + s1 * auxA[n]
                   + s2 * auxA[(n + 1) & (NROWS - 1)];
    wv[j] = powf(wh, gamma);
    psum += wv[j];
  }
  const float tsum = block_reduce_sum(psum, wscr, lane, wid);
  const float winv = 1.f / (tsum + EPSF);
  #pragma unroll
  for (int j = 0; j < 4; ++j) dotA[tid + 256 * j] = wv[j] * winv;
  __syncthreads();   // final w visible to all

  // ---- stage 4: read[m] = sum_n w[n] * memory[n][m]  (per-wave partial GEMV) ----
  {
    float r0 = 0.f, r1 = 0.f;
    const int base = wid * 128;
    for (int n = 0; n < 128; ++n) {
      const float wn = dotA[base + n];
      r0 += wn * tile[(base + n) * PITCH + lane];
      r1 += wn * tile[(base + n) * PITCH + lane + 32];
    }
    atomicAdd(&racc[lane],      r0);   // ds_add_f32
    atomicAdd(&racc[lane + 32], r1);
  }
  __syncthreads();

  // ---- stage 5: emit read row + erase/add update, single HBM write pass ----
  float* outB = out + (size_t)b * (NROWS + 1) * MCOLS;
  if (tid < 64) outB[tid] = racc[tid];

  for (int idx = tid; idx < NROWS * MCOLS / 4; idx += 256) {
    const int row = idx >> 4;
    const int c   = (idx & 15) * 4;
    const float wn = dotA[row];
    float4 mv = *(const float4*)(tile + row * PITCH + c);
    float4 ov;
    ov.x = mv.x * (1.f - wn * eb[c + 0]) + wn * ab[c + 0];
    ov.y = mv.y * (1.f - wn * eb[c + 1]) + wn * ab[c + 1];
    ov.z = mv.z * (1.f - wn * eb[c + 2]) + wn * ab[c + 2];
    ov.w = mv.w * (1.f - wn * eb[c + 3]) + wn * ab[c + 3];
    *(float4*)(outB + (size_t)(row + 1) * MCOLS + c) = ov;
  }
}

extern "C" void kernel_launch(void* const* d_in, const int* in_sizes, int n_in,
                              void* d_out, int out_size, void* d_ws, size_t ws_size,
                              hipStream_t stream) {
  (void)in_sizes; (void)n_in; (void)out_size; (void)d_ws; (void)ws_size;
  const float* memory = (const float*)d_in[0];
  const float* k      = (const float*)d_in[1];
  const float* beta   = (const float*)d_in[2];
  const float* g      = (const float*)d_in[3];
  const float* s      = (const float*)d_in[4];
  const float* gamma  = (const float*)d_in[5];
  const float* w_prev = (const float*)d_in[6];
  const float* e      = (const float*)d_in[7];
  const float* a      = (const float*)d_in[8];
  float* out = (float*)d_out;

  const size_t lds_bytes =
      (size_t)(NROWS * PITCH + NROWS + NROWS + 64 * 4 + 32) * sizeof(float); // 287872 B < 320KB
  hipFuncSetAttribute((const void*)ntm_memory_kernel,
                      hipFuncAttributeMaxDynamicSharedMemorySize, (int)lds_bytes);

  ntm_memory_kernel<<<NBATCH, 256, lds_bytes, stream>>>(
      memory, k, beta, g, s, gamma, w_prev, e, a, out);
}